// AnimationNet_20229295964864
// MI455X (gfx1250) — compile-verified
//
#include <hip/hip_runtime.h>
#include <math.h>

// Shapes from reference
#define BB 8
#define PP 8
#define NN 27554
#define JJ 24
#define TN (3*NN)          // 82662 columns of posedirs
#define KF 207             // pose feature length
#define KFP 208            // padded to multiple of 4
#define NTB 4              // N-tiles per block in pose GEMM (64 columns)

typedef __attribute__((ext_vector_type(2))) float v2f;
typedef __attribute__((ext_vector_type(8))) float v8f;

__constant__ int c_parents[24] = {-1,0,0,0,1,2,3,4,5,6,7,8,9,9,9,12,13,14,16,17,18,19,20,21};

// ---------------------------------------------------------------------------
// Kernel A: Rodrigues + pose_feature (transposed) + FK chain -> A matrices
// grid = 64 blocks (one per (b,p)), 32 threads (one wave)
// PFT : [208][64]  pose_feature transposed, row 207 zero padded
// Bmat: [8][24][96] per-b WMMA B operand: row j, col p*12+e, e=r*4+c of A[:3,:4]
// ---------------------------------------------------------------------------
__global__ __launch_bounds__(32) void ker_prep(const float* __restrict__ body_pose,
                                               const float* __restrict__ joints,
                                               float* __restrict__ PFT,
                                               float* __restrict__ Bmat)
{
    const int bp = blockIdx.x;          // b*8+p
    const int b  = bp >> 3;
    const int p  = bp & 7;
    const int lane = threadIdx.x;

    __shared__ float sTm[24][12];       // [R | rel_t] row-major 3x4
    __shared__ float sCh[24][12];       // chain transforms

    if (lane < 24) {
        const int j = lane;
        const float rv0 = body_pose[(size_t)bp*72 + j*3 + 0];
        const float rv1 = body_pose[(size_t)bp*72 + j*3 + 1];
        const float rv2 = body_pose[(size_t)bp*72 + j*3 + 2];
        // angle = ||rv + 1e-8|| (componentwise eps, per reference)
        const float e0 = rv0 + 1e-8f, e1 = rv1 + 1e-8f, e2 = rv2 + 1e-8f;
        const float angle = sqrtf(e0*e0 + e1*e1 + e2*e2);
        const float kx = rv0 / angle, ky = rv1 / angle, kz = rv2 / angle;
        const float s = sinf(angle), c = cosf(angle), oc = 1.0f - c;
        const float R00 = 1.0f - oc*(ky*ky + kz*kz);
        const float R01 = -s*kz + oc*(kx*ky);
        const float R02 =  s*ky + oc*(kx*kz);
        const float R10 =  s*kz + oc*(kx*ky);
        const float R11 = 1.0f - oc*(kx*kx + kz*kz);
        const float R12 = -s*kx + oc*(ky*kz);
        const float R20 = -s*ky + oc*(kx*kz);
        const float R21 =  s*kx + oc*(ky*kz);
        const float R22 = 1.0f - oc*(kx*kx + ky*ky);

        const int par = c_parents[j];
        float rx = joints[j*3+0], ry = joints[j*3+1], rz = joints[j*3+2];
        if (par >= 0) { rx -= joints[par*3+0]; ry -= joints[par*3+1]; rz -= joints[par*3+2]; }

        sTm[j][0]=R00; sTm[j][1]=R01; sTm[j][2] =R02; sTm[j][3] =rx;
        sTm[j][4]=R10; sTm[j][5]=R11; sTm[j][6] =R12; sTm[j][7] =ry;
        sTm[j][8]=R20; sTm[j][9]=R21; sTm[j][10]=R22; sTm[j][11]=rz;

        if (j >= 1) {   // pose_feature = (R - I), joints 1..23, transposed store
            const int f0 = (j-1)*9;
            PFT[(f0+0)*64 + bp] = R00 - 1.0f;
            PFT[(f0+1)*64 + bp] = R01;
            PFT[(f0+2)*64 + bp] = R02;
            PFT[(f0+3)*64 + bp] = R10;
            PFT[(f0+4)*64 + bp] = R11 - 1.0f;
            PFT[(f0+5)*64 + bp] = R12;
            PFT[(f0+6)*64 + bp] = R20;
            PFT[(f0+7)*64 + bp] = R21;
            PFT[(f0+8)*64 + bp] = R22 - 1.0f;
        }
    }
    if (lane == 0) PFT[207*64 + bp] = 0.0f;   // K pad
    __syncthreads();

    if (lane == 0) {   // serial FK chain (23 affine 3x4 multiplies)
        #pragma unroll 1
        for (int j = 0; j < 24; ++j) {
            if (j == 0) {
                #pragma unroll
                for (int i = 0; i < 12; ++i) sCh[0][i] = sTm[0][i];
            } else {
                const int par = c_parents[j];
                float Pm[12], Tm[12];
                #pragma unroll
                for (int i = 0; i < 12; ++i) { Pm[i] = sCh[par][i]; Tm[i] = sTm[j][i]; }
                #pragma unroll
                for (int r = 0; r < 3; ++r) {
                    #pragma unroll
                    for (int cc = 0; cc < 4; ++cc) {
                        float v = Pm[r*4+0]*Tm[0*4+cc] + Pm[r*4+1]*Tm[1*4+cc] + Pm[r*4+2]*Tm[2*4+cc];
                        if (cc == 3) v += Pm[r*4+3];
                        sCh[j][r*4+cc] = v;
                    }
                }
            }
        }
    }
    __syncthreads();

    if (lane < 24) {   // A[j] = [chainR | chain_t - chainR @ joints[j]]
        const int j = lane;
        float M[12];
        #pragma unroll
        for (int i = 0; i < 12; ++i) M[i] = sCh[j][i];
        const float jx = joints[j*3+0], jy = joints[j*3+1], jz = joints[j*3+2];
        M[3]  -= M[0]*jx + M[1]*jy + M[2] *jz;
        M[7]  -= M[4]*jx + M[5]*jy + M[6] *jz;
        M[11] -= M[8]*jx + M[9]*jy + M[10]*jz;
        float* dst = Bmat + (size_t)b*(24*96) + j*96 + p*12;
        #pragma unroll
        for (int i = 0; i < 12; ++i) dst[i] = M[i];
    }
}

// ---------------------------------------------------------------------------
// Kernel B: pose_offsets = PF(64x207) @ posedirs(207x82662) via WMMA fp32
// grid = ceil(TN/64) blocks x 128 threads; wave w handles M-tile w.
// Each block covers 4 N-tiles: A fragment loaded once per K-step and reused
// across 4 independent accumulators (no WMMA RAW hazards, 4x less PF traffic).
// posedirs panel staged in LDS as [208][64] row-major (coalesced fill).
// ---------------------------------------------------------------------------
__global__ __launch_bounds__(128) void ker_posegemm(const float* __restrict__ PFT,
                                                    const float* __restrict__ pd,
                                                    float* __restrict__ po)
{
    const int tid  = threadIdx.x;
    const int w    = tid >> 5;
    const int lane = tid & 31;
    const int half = lane >> 4;
    const int lm   = lane & 15;
    const int n0   = blockIdx.x * (16*NTB);

    __shared__ float sBp[KFP*64];   // B panel: 208 rows x 64 cols (row 207 = 0)
    for (int idx = tid; idx < KFP*64; idx += 128) {
        const int row = idx >> 6, col = idx & 63;
        const int n = n0 + col;
        sBp[idx] = (row < KF && n < TN) ? pd[(size_t)row*TN + n] : 0.0f;
    }
    __syncthreads();

    const int m = w*16 + lm;        // bp row (0..63)
    v8f acc[NTB];
    #pragma unroll
    for (int nt = 0; nt < NTB; ++nt)
        acc[nt] = (v8f){0.f,0.f,0.f,0.f,0.f,0.f,0.f,0.f};

    #pragma unroll 2
    for (int ks = 0; ks < 52; ++ks) {
        const int kk = ks*4 + half*2;
        v2f a;
        a.x = PFT[(kk  )*64 + m];
        a.y = PFT[(kk+1)*64 + m];
        #pragma unroll
        for (int nt = 0; nt < NTB; ++nt) {
            v2f bf;
            bf.x = sBp[(kk  )*64 + nt*16 + lm];
            bf.y = sBp[(kk+1)*64 + nt*16 + lm];
            acc[nt] = __builtin_amdgcn_wmma_f32_16x16x4_f32(false, a, false, bf,
                                                            (short)0, acc[nt], false, false);
        }
    }

    #pragma unroll
    for (int nt = 0; nt < NTB; ++nt) {
        const int n = n0 + nt*16 + lm;
        if (n < TN) {
            #pragma unroll
            for (int r = 0; r < 8; ++r) {
                const int mm = r + 8*half;          // row inside 16-row tile
                po[(size_t)(w*16 + mm)*TN + n] = acc[nt][r];
            }
        }
    }
}

// ---------------------------------------------------------------------------
// Kernel C: fused T-einsum (WMMA) + LBS transform + quaternion math
// grid = (ceil(N/128), B) x 256 threads (8 waves, 16 points per wave)
// ---------------------------------------------------------------------------
__global__ __launch_bounds__(256) void ker_lbs(const float* __restrict__ xyz,
                                               const float* __restrict__ rot,
                                               const float* __restrict__ lbsw,
                                               const float* __restrict__ trans,
                                               const float* __restrict__ Bmat,
                                               const float* __restrict__ po,
                                               float* __restrict__ out)
{
    const int b    = blockIdx.y;
    const int tid  = threadIdx.x;
    const int w    = tid >> 5;
    const int lane = tid & 31;
    const int half = lane >> 4;
    const int lm   = lane & 15;
    const int base = blockIdx.x*128 + w*16;

    __shared__ float sB[24*96];          // per-b A matrices (WMMA B operand)
    __shared__ float sT[8][16][100];     // per-wave T tiles: 16 pts x 96 (pad 100)

    for (int idx = tid; idx < 24*96; idx += 256)
        sB[idx] = Bmat[(size_t)b*(24*96) + idx];
    __syncthreads();

    // A fragments: lbs_weights rows for this wave's 16 points, K=24 in 6 steps
    v2f afr[6];
    {
        const int pt = base + lm;
        const bool ok = pt < NN;
        const float* wr = lbsw + ((size_t)b*NN + (ok ? pt : 0))*JJ;
        #pragma unroll
        for (int ks = 0; ks < 6; ++ks) {
            const int kk = ks*4 + half*2;
            afr[ks].x = ok ? wr[kk]   : 0.0f;
            afr[ks].y = ok ? wr[kk+1] : 0.0f;
        }
    }

    // T tile = weights(16x24) @ A(24x96): 6 N-tiles x 6 K-steps of WMMA fp32
    #pragma unroll 1
    for (int nt = 0; nt < 6; ++nt) {
        v8f acc = {0.f,0.f,0.f,0.f,0.f,0.f,0.f,0.f};
        #pragma unroll
        for (int ks = 0; ks < 6; ++ks) {
            const int kk = ks*4 + half*2;
            v2f bf;
            bf.x = sB[(kk  )*96 + nt*16 + lm];
            bf.y = sB[(kk+1)*96 + nt*16 + lm];
            acc = __builtin_amdgcn_wmma_f32_16x16x4_f32(false, afr[ks], false, bf,
                                                        (short)0, acc, false, false);
        }
        #pragma unroll
        for (int r = 0; r < 8; ++r)
            sT[w][r + 8*half][nt*16 + lm] = acc[r];
    }
    __syncthreads();

    // Per-point epilogue: lane -> point lm, p in [half*4, half*4+4)
    const int pt = base + lm;
    if (pt < NN) {
        const float x = xyz[((size_t)b*NN + pt)*3 + 0];
        const float y = xyz[((size_t)b*NN + pt)*3 + 1];
        const float z = xyz[((size_t)b*NN + pt)*3 + 2];
        const float qr = rot[((size_t)b*NN + pt)*4 + 0];
        const float qi = rot[((size_t)b*NN + pt)*4 + 1];
        const float qj = rot[((size_t)b*NN + pt)*4 + 2];
        const float qk = rot[((size_t)b*NN + pt)*4 + 3];
        const float two_s = 2.0f / (qr*qr + qi*qi + qj*qj + qk*qk);
        float Rm[3][3];
        Rm[0][0] = 1.0f - two_s*(qj*qj + qk*qk);
        Rm[0][1] = two_s*(qi*qj - qk*qr);
        Rm[0][2] = two_s*(qi*qk + qj*qr);
        Rm[1][0] = two_s*(qi*qj + qk*qr);
        Rm[1][1] = 1.0f - two_s*(qi*qi + qk*qk);
        Rm[1][2] = two_s*(qj*qk - qi*qr);
        Rm[2][0] = two_s*(qi*qk - qj*qr);
        Rm[2][1] = two_s*(qj*qk + qi*qr);
        Rm[2][2] = 1.0f - two_s*(qi*qi + qj*qj);

        #pragma unroll 1
        for (int dp = 0; dp < 4; ++dp) {
            const int p = half*4 + dp;
            float T[12];
            #pragma unroll
            for (int i = 0; i < 12; ++i) T[i] = sT[w][lm][p*12 + i];

            const size_t bp = (size_t)b*PP + p;
            const float po0 = po[bp*TN + (size_t)pt*3 + 0];
            const float po1 = po[bp*TN + (size_t)pt*3 + 1];
            const float po2 = po[bp*TN + (size_t)pt*3 + 2];
            const float px = x + po0, py = y + po1, pz = z + po2;

            const float xt0 = T[0]*px + T[1]*py + T[2] *pz + T[3]  + trans[bp*3+0];
            const float xt1 = T[4]*px + T[5]*py + T[6] *pz + T[7]  + trans[bp*3+1];
            const float xt2 = T[8]*px + T[9]*py + T[10]*pz + T[11] + trans[bp*3+2];

            float Nw[3][3];
            #pragma unroll
            for (int r = 0; r < 3; ++r)
                #pragma unroll
                for (int cc = 0; cc < 3; ++cc)
                    Nw[r][cc] = T[r*4+0]*Rm[0][cc] + T[r*4+1]*Rm[1][cc] + T[r*4+2]*Rm[2][cc];

            const float m00=Nw[0][0], m01=Nw[0][1], m02=Nw[0][2];
            const float m10=Nw[1][0], m11=Nw[1][1], m12=Nw[1][2];
            const float m20=Nw[2][0], m21=Nw[2][1], m22=Nw[2][2];
            float qa0 = sqrtf(fmaxf(1.0f + m00 + m11 + m22, 0.0f));
            float qa1 = sqrtf(fmaxf(1.0f + m00 - m11 - m22, 0.0f));
            float qa2 = sqrtf(fmaxf(1.0f - m00 + m11 - m22, 0.0f));
            float qa3 = sqrtf(fmaxf(1.0f - m00 - m11 + m22, 0.0f));
            int best = 0; float bv = qa0;
            if (qa1 > bv) { bv = qa1; best = 1; }
            if (qa2 > bv) { bv = qa2; best = 2; }
            if (qa3 > bv) { bv = qa3; best = 3; }
            float c0, c1, c2, c3;
            if      (best == 0) { c0 = qa0*qa0;  c1 = m21-m12;  c2 = m02-m20;  c3 = m10-m01; }
            else if (best == 1) { c0 = m21-m12;  c1 = qa1*qa1;  c2 = m10+m01;  c3 = m02+m20; }
            else if (best == 2) { c0 = m02-m20;  c1 = m10+m01;  c2 = qa2*qa2;  c3 = m12+m21; }
            else                { c0 = m10-m01;  c1 = m20+m02;  c2 = m21+m12;  c3 = qa3*qa3; }
            const float inv = 1.0f / (2.0f * fmaxf(bv, 0.1f));

            float* op = out + (bp*(size_t)NN + pt)*7;
            op[0] = xt0; op[1] = xt1; op[2] = xt2;
            op[3] = c0*inv; op[4] = c1*inv; op[5] = c2*inv; op[6] = c3*inv;
        }
    }
}

// ---------------------------------------------------------------------------
extern "C" void kernel_launch(void* const* d_in, const int* in_sizes, int n_in,
                              void* d_out, int out_size, void* d_ws, size_t ws_size,
                              hipStream_t stream) {
    const float* xyz       = (const float*)d_in[0];
    const float* rot       = (const float*)d_in[1];
    const float* lbsw      = (const float*)d_in[2];
    const float* body_pose = (const float*)d_in[3];
    const float* trans     = (const float*)d_in[4];
    const float* posedirs  = (const float*)d_in[5];
    const float* joints    = (const float*)d_in[6];
    float* out = (float*)d_out;

    float* ws   = (float*)d_ws;
    float* PFT  = ws;                       // 208*64 floats
    float* Bmat = PFT + KFP*64;             // 8*24*96 floats
    float* po   = Bmat + (size_t)BB*24*96;  // 64 * 82662 floats (~21 MB)

    ker_prep<<<BB*PP, 32, 0, stream>>>(body_pose, joints, PFT, Bmat);
    ker_posegemm<<<(TN + 16*NTB - 1)/(16*NTB), 128, 0, stream>>>(PFT, posedirs, po);
    dim3 g((NN + 127)/128, BB);
    ker_lbs<<<g, 256, 0, stream>>>(xyz, rot, lbsw, trans, Bmat, po, out);
}